// DepthToSpace_678604833108
// MI455X (gfx1250) — compile-verified
//
#include <hip/hip_runtime.h>

// DepthToSpace, block=2, x: (16, 256, 128, 128) f32 -> out: (16, 64, 256, 256) f32
// out[b, dd, 2h+i, 2w+k] = x[b, (2i+k)*64 + dd, h, w]
//
// Pure bandwidth-bound permutation (512 MB total traffic, ~22 us at 23.3 TB/s).
// Strategy: one float4 of output per thread; two coalesced b64 NT loads
// (channels c0, c0+64) + one coalesced b128 NT store. NT temporal hints keep
// the 512 MB stream from thrashing the 192 MB L2.

typedef __attribute__((ext_vector_type(2))) float v2f;
typedef __attribute__((ext_vector_type(4))) float v4f;

__global__ __launch_bounds__(256) void depth_to_space_kernel(
    const float* __restrict__ x, float* __restrict__ out, unsigned total4) {
  unsigned idx = blockIdx.x * 256u + threadIdx.x;
  if (idx >= total4) return;

  // idx -> (b, dd, ho, q) with q = wo/4 in [0,64), ho in [0,256), dd in [0,64), b in [0,16)
  unsigned q  = idx & 63u;
  unsigned t  = idx >> 6;
  unsigned ho = t & 255u;
  t >>= 8;
  unsigned dd = t & 63u;
  unsigned b  = t >> 6;

  unsigned h = ho >> 1;
  unsigned i = ho & 1u;
  unsigned c0 = i * 128u + dd;               // channel for even wo; c1 = c0 + 64

  // x[b, c0, h, 2q]
  size_t in_off = (((size_t)b * 256u + c0) * 128u + h) * 128u + (q * 2u);
  const v2f* p0 = (const v2f*)(x + in_off);
  const v2f* p1 = (const v2f*)(x + in_off + (size_t)64u * 128u * 128u);

  v2f a0 = __builtin_nontemporal_load(p0);   // w = 2q, 2q+1 from c0
  v2f a1 = __builtin_nontemporal_load(p1);   // w = 2q, 2q+1 from c1

  v4f o;
  o.x = a0.x;   // wo = 4q   (k=0)
  o.y = a1.x;   // wo = 4q+1 (k=1)
  o.z = a0.y;   // wo = 4q+2 (k=0)
  o.w = a1.y;   // wo = 4q+3 (k=1)

  // out[b, dd, ho, 4q]
  size_t out_off = (((size_t)b * 64u + dd) * 256u + ho) * 256u + (q * 4u);
  __builtin_nontemporal_store(o, (v4f*)(out + out_off));
}

extern "C" void kernel_launch(void* const* d_in, const int* in_sizes, int n_in,
                              void* d_out, int out_size, void* d_ws, size_t ws_size,
                              hipStream_t stream) {
  (void)in_sizes; (void)n_in; (void)d_ws; (void)ws_size;
  const float* x = (const float*)d_in[0];
  float* out = (float*)d_out;

  unsigned total4 = (unsigned)(out_size / 4);          // 16,777,216 float4s
  unsigned blocks = (total4 + 255u) / 256u;            // 65,536 blocks
  depth_to_space_kernel<<<blocks, 256, 0, stream>>>(x, out, total4);
}